// NonlinearOscillatorNet_7284264534176
// MI455X (gfx1250) — compile-verified
//
#include <hip/hip_runtime.h>
#include <hip/hip_bf16.h>

typedef __attribute__((ext_vector_type(2))) float v2f;
typedef __attribute__((ext_vector_type(4))) float v4f;
typedef __attribute__((ext_vector_type(8))) float v8f;

#define NOSC 512
#define BATCH 256
#define TSTEPS 200
#define DT 0.1f

// Packed-layout geometry
//   Kpack : [slab(128)][ntile(32)][lane(32)] float4 = {Kr[j,n],Kr[j+1,n],Ki[j,n],Ki[j+1,n]}
//           with j = slab*4 + (lane>=16 ? 2:0), n = ntile*16 + (lane&15)
//   ZApack: [mtile(16)][slab(128)][lane(32)] float4 = {Zr[row,j],Zr[row,j+1],Zi[row,j],Zi[row,j+1]}
//           with row = mtile*16 + (lane&15), j as above
// Both match the V_WMMA_F32_16X16X4_F32 A/B fragment layouts exactly:
// one b128 load per lane per operand per slab.

// ---------------------------------------------------------------------------
// Prep: pack K into WMMA-B fragments, broadcast z_init into WMMA-A fragments.
// ---------------------------------------------------------------------------
__global__ __launch_bounds__(256)
void osc_prep_kernel(const float* __restrict__ Kc,      // (N,N) c64 interleaved
                     const float* __restrict__ z_init,  // (1,N) c64 interleaved
                     v4f* __restrict__ Kpack,           // 128*32*32 float4
                     v4f* __restrict__ ZA0)             // 16*128*32 float4
{
    const int i = blockIdx.x * blockDim.x + threadIdx.x;
    if (i < 128 * 32 * 32) {
        const int lane  = i & 31;
        const int ntile = (i >> 5) & 31;
        const int slab  = i >> 10;
        const int n  = ntile * 16 + (lane & 15);
        const int j0 = slab * 4 + ((lane >> 4) << 1);
        v4f v;
        v.x = Kc[2 * (j0 * NOSC + n)];            // Kr[j0 , n]
        v.y = Kc[2 * ((j0 + 1) * NOSC + n)];      // Kr[j0+1, n]
        v.z = Kc[2 * (j0 * NOSC + n) + 1];        // Ki[j0 , n]
        v.w = Kc[2 * ((j0 + 1) * NOSC + n) + 1];  // Ki[j0+1, n]
        Kpack[i] = v;
    }
    if (i < 16 * 128 * 32) {
        const int lane = i & 31;
        const int slab = (i >> 5) & 127;
        const int j0 = slab * 4 + ((lane >> 4) << 1);
        v4f v;                                     // broadcast over rows
        v.x = z_init[2 * j0];
        v.y = z_init[2 * (j0 + 1)];
        v.z = z_init[2 * j0 + 1];
        v.w = z_init[2 * (j0 + 1) + 1];
        ZA0[i] = v;
    }
}

// ---------------------------------------------------------------------------
// One recurrence step. One wave per 16x16 coupling tile; 128 blocks x 128 thr
// = 512 waves. Software-pipelined (prefetch depth 2): per slab, 2 coalesced
// b128 loads feed 4 f32 WMMA chains (RR,II,RI,IR). Fused oscillator update.
// ---------------------------------------------------------------------------
#define OSC_WMMA4(av, bv)                                                     \
    do {                                                                      \
        const v2f ar_ = {(av).x, (av).y};                                     \
        const v2f ai_ = {(av).z, (av).w};                                     \
        const v2f br_ = {(bv).x, (bv).y};                                     \
        const v2f bi_ = {(bv).z, (bv).w};                                     \
        accRR = __builtin_amdgcn_wmma_f32_16x16x4_f32(false, ar_, false, br_, \
                                                      (short)0, accRR, false, false); \
        accII = __builtin_amdgcn_wmma_f32_16x16x4_f32(false, ai_, false, bi_, \
                                                      (short)0, accII, false, false); \
        accRI = __builtin_amdgcn_wmma_f32_16x16x4_f32(false, ar_, false, bi_, \
                                                      (short)0, accRI, false, false); \
        accIR = __builtin_amdgcn_wmma_f32_16x16x4_f32(false, ai_, false, br_, \
                                                      (short)0, accIR, false, false); \
    } while (0)

__global__ __launch_bounds__(128)
void osc_step_kernel(const v4f* __restrict__ Kpack,
                     const v4f* __restrict__ ZA_in,
                     v4f* __restrict__ ZA_out,
                     const float* __restrict__ I_t_seq,   // (B,1,T) f32
                     const float* __restrict__ mu,        // (1,N) f32
                     const float* __restrict__ omega,     // (1,N) f32
                     const float* __restrict__ Q,         // (N,1) c64
                     const float* __restrict__ r_param,   // (1,N) c64
                     float* __restrict__ traj,            // (B,T,N) c64
                     int t)
{
    const int lane  = threadIdx.x & 31;
    const int wave  = threadIdx.x >> 5;
    const int tile  = blockIdx.x * (blockDim.x >> 5) + wave;  // 0..511
    const int mtile = tile >> 5;    // 0..15
    const int ntile = tile & 31;    // 0..31
    const int m0 = mtile << 4;
    const int n0 = ntile << 4;
    const int half = lane >> 4;
    const int l16  = lane & 15;

    v8f accRR = {}; v8f accII = {}; v8f accRI = {}; v8f accIR = {};

    const v4f* Ap = ZA_in + (size_t)mtile * (128 * 32) + lane;  // +32 per slab
    const v4f* Bp = Kpack + (size_t)ntile * 32 + lane;          // +1024 per slab

    // ---- software-pipelined GEMM loop, prefetch depth 2 -------------------
    v4f a0 = Ap[0];
    v4f b0 = Bp[0];
    v4f a1 = Ap[32];
    v4f b1 = Bp[1024];

    #pragma unroll 2
    for (int slab = 0; slab < 126; ++slab) {
        const v4f an = Ap[(slab + 2) * 32];
        const v4f bn = Bp[(size_t)(slab + 2) * 1024];
        OSC_WMMA4(a0, b0);
        a0 = a1; b0 = b1;
        a1 = an; b1 = bn;
    }
    OSC_WMMA4(a0, b0);
    OSC_WMMA4(a1, b1);

    // ---- fused oscillator update (elementwise in (b,n)) -------------------
    const int   n   = n0 + l16;
    const float qr  = Q[2 * n],       qi = Q[2 * n + 1];
    const float rr  = r_param[2 * n], ri = r_param[2 * n + 1];
    const float mun = mu[n];
    const float omn = omega[n];

    const float* Zin_f  = (const float*)ZA_in;
    float*       Zout_f = (float*)ZA_out;
    // packed-A float index of element (b, n): base + (b&15)*4 ; imag at +2
    const int base = (((mtile * 128 + (n >> 2)) * 32 + (((n >> 1) & 1) << 4)) << 2)
                     + (n & 1);

    #pragma unroll
    for (int v = 0; v < 8; ++v) {
        // C/D 16x16 f32 layout: VGPR v -> M=v (lanes 0-15) / M=v+8 (lanes 16-31)
        const int b15 = v + (half << 3);
        const int b   = m0 + b15;
        const int idx = base + (b15 << 2);

        const float cr = accRR[v] - accII[v];
        const float ci = accRI[v] + accIR[v];

        const float zr = Zin_f[idx];
        const float zi = Zin_f[idx + 2];
        const float zrr = zr - rr;
        const float zri = zi - ri;
        const float mag2 = zrr * zrr + zri * zri;

        const float fr = mun + cr - mag2;    // Re(lam + coupling - |z-r|^2)
        const float fi = omn + ci;           // Im(...)

        const float It  = I_t_seq[b * TSTEPS + t];
        const float dzr = fr * zrr - fi * zri + qr * It;
        const float dzi = fr * zri + fi * zrr + qi * It;

        const float znr = zr + DT * dzr;
        const float zni = zi + DT * dzi;

        Zout_f[idx]     = znr;
        Zout_f[idx + 2] = zni;

        const size_t ti = ((size_t)(b * TSTEPS + t) * NOSC + n) * 2;
        traj[ti]     = znr;
        traj[ti + 1] = zni;
    }
}

// ---------------------------------------------------------------------------
// Readout: o[b,t] = |sum_n z_traj[b,t,n] * D[n]|^2. One wave per (b,t).
// ---------------------------------------------------------------------------
__global__ __launch_bounds__(256)
void osc_out_kernel(const float* __restrict__ traj,
                    const float* __restrict__ Dm,
                    float* __restrict__ o)
{
    const int gwave = (blockIdx.x * blockDim.x + threadIdx.x) >> 5;  // b*T + t
    const int lane  = threadIdx.x & 31;
    if (gwave >= BATCH * TSTEPS) return;

    const float* zp = traj + (size_t)gwave * NOSC * 2;
    float sr = 0.0f, si = 0.0f;
    #pragma unroll 4
    for (int n = lane; n < NOSC; n += 32) {
        const float zr = zp[2 * n], zi = zp[2 * n + 1];
        const float dr = Dm[2 * n], di = Dm[2 * n + 1];
        sr += zr * dr - zi * di;
        si += zr * di + zi * dr;
    }
    #pragma unroll
    for (int off = 16; off > 0; off >>= 1) {
        sr += __shfl_xor(sr, off, 32);
        si += __shfl_xor(si, off, 32);
    }
    if (lane == 0) o[gwave] = sr * sr + si * si;
}

// ---------------------------------------------------------------------------
extern "C" void kernel_launch(void* const* d_in, const int* in_sizes, int n_in,
                              void* d_out, int out_size, void* d_ws, size_t ws_size,
                              hipStream_t stream)
{
    const float* I_t_seq = (const float*)d_in[0];
    const float* mu      = (const float*)d_in[1];
    const float* omega   = (const float*)d_in[2];
    const float* Kc      = (const float*)d_in[3];
    const float* Q       = (const float*)d_in[4];
    const float* r_param = (const float*)d_in[5];
    const float* z_init  = (const float*)d_in[6];
    const float* Dm      = (const float*)d_in[7];

    float* out  = (float*)d_out;
    float* o    = out;                       // (B,T) = 51200 floats
    float* traj = out + BATCH * TSTEPS;      // (B,T,N) complex as float pairs

    // workspace (floats): Kpack (524288) | ZA0 (262144) | ZA1 (262144) = 4 MB
    float* ws  = (float*)d_ws;
    v4f* Kpack = (v4f*)ws;
    v4f* ZA0   = (v4f*)(ws + 524288);
    v4f* ZA1   = (v4f*)(ws + 524288 + 262144);

    osc_prep_kernel<<<(128 * 32 * 32 + 255) / 256, 256, 0, stream>>>(
        Kc, z_init, Kpack, ZA0);

    for (int t = 0; t < TSTEPS; ++t) {
        const bool odd = (t & 1) != 0;
        const v4f* zin  = odd ? ZA1 : ZA0;
        v4f*       zout = odd ? ZA0 : ZA1;
        osc_step_kernel<<<128, 128, 0, stream>>>(
            Kpack, zin, zout, I_t_seq, mu, omega, Q, r_param, traj, t);
    }

    const int nwaves = BATCH * TSTEPS;                 // 51200
    osc_out_kernel<<<(nwaves * 32 + 255) / 256, 256, 0, stream>>>(traj, Dm, o);
}